// KarateGraph5Att_68599217652371
// MI455X (gfx1250) — compile-verified
//
#include <hip/hip_runtime.h>
#include <math.h>

typedef __attribute__((ext_vector_type(2))) float v2f;
typedef __attribute__((ext_vector_type(8))) float v8f;
typedef __attribute__((ext_vector_type(4))) unsigned int u32x4;
typedef __attribute__((ext_vector_type(4))) int i32x4;
typedef __attribute__((ext_vector_type(8))) int i32x8;

#define GAT_NEG_SLOPE 0.2f

#if __has_builtin(__builtin_amdgcn_tensor_load_to_lds)
#define USE_TDM 1
#else
#define USE_TDM 0
#endif

// Monotonic float<->int encoding so integer atomicMax == float max (handles negatives, -inf).
__device__ __forceinline__ int fenc(float f) {
  int i = __float_as_int(f);
  return (i < 0) ? (i ^ 0x7FFFFFFF) : i;
}
__device__ __forceinline__ float fdec(int i) {
  int j = (i < 0) ? (i ^ 0x7FFFFFFF) : i;
  return __int_as_float(j);
}

// ---------------------------------------------------------------------------
// Tiled WMMA GEMM: H[N,D] = X[N,K] @ W[K,D], fp32 16x16x4 WMMA.
// Block = 128 threads (4 waves) -> 64x64 output tile (wave w: rows w*16..+16,
// all 64 cols as 4 accumulators). W K-chunks (KBLK x 64) staged in LDS by the
// Tensor Data Mover (one 16-row slice per wave), double buffered, padded to an
// 80-float row stride to avoid DS bank conflicts. A is streamed as contiguous
// b64 loads per wave. Requires K%KBLK==0, D%64==0.
// ---------------------------------------------------------------------------
#define KBLK 64
#define LDSTR 80                    // padded LDS row stride (floats)
#define BUF_FLOATS (KBLK * LDSTR)   // 5120 floats = 20480 B per buffer

#if USE_TDM
// DMA 16 x 64 fp32 tile from gsrc (row stride D elems) into LDS at byte
// offset lds_off, inserting 16 pad dwords after every 64 dwords (row stride 80).
__device__ __forceinline__ void tdm_load_w(const float* gsrc, unsigned int lds_off, int D) {
  unsigned long long ga = (unsigned long long)(size_t)gsrc;
  u32x4 g0;
  g0.x = 1u;                                                  // count=1, user D#
  g0.y = lds_off;                                             // lds_addr (bytes)
  g0.z = (unsigned int)ga;                                    // global_addr[31:0]
  g0.w = (unsigned int)((ga >> 32) & 0x1FFFFFFull) | (2u << 30);  // addr[56:32] | type=2
  unsigned int td0 = (unsigned int)D;   // tensor_dim0 (elems per row)
  unsigned int td1 = 16u;               // tensor_dim1 (rows we touch)
  i32x8 g1;
  g1[0] = (int)((2u << 16) | (1u << 20) | (5u << 22) | (15u << 25));
          // data_size=4B | pad_enable | pad_interval=64dw | pad_amount=16dw
  g1[1] = (int)((td0 & 0xFFFFu) << 16);                       // [63:48]=td0.lo
  g1[2] = (int)((td0 >> 16) | ((td1 & 0xFFFFu) << 16));       // td0.hi | td1.lo
  g1[3] = (int)((td1 >> 16) | (64u << 16));                   // td1.hi | tile_dim0=64
  g1[4] = 16;                                                 // tile_dim1=16, tile_dim2=0
  g1[5] = (int)(unsigned int)D;                               // tensor_dim0_stride lo32
  g1[6] = 0;                                                  // stride hi16 | dim1_stride lo
  g1[7] = 0;
  i32x4 gz = {0, 0, 0, 0};
#if __clang_major__ >= 23
  i32x8 gz8 = {};
  __builtin_amdgcn_tensor_load_to_lds(g0, g1, gz, gz, gz8, 0);
#else
  __builtin_amdgcn_tensor_load_to_lds(g0, g1, gz, gz, 0);
#endif
}
#endif

__global__ void __launch_bounds__(128)
gemm_wmma_tile(const float* __restrict__ X, const float* __restrict__ W,
               float* __restrict__ H, int N, int K, int D) {
  __shared__ float lds_w[2 * BUF_FLOATS];

  const int lane = threadIdx.x & 31;
  const int wave = __builtin_amdgcn_readfirstlane((int)(threadIdx.x >> 5));
  const int r = lane & 15;
  const bool hi = lane >= 16;
  const int koff = hi ? 2 : 0;
  const int n0 = blockIdx.y * 64;

  int mbase = blockIdx.x * 64 + wave * 16;   // clamp tail block: duplicate work,
  if (mbase > N - 16) mbase = N - 16;        // identical values written twice.

  const float* __restrict__ xrow = X + (size_t)(mbase + r) * K;
  const int C = K / KBLK;
  v8f acc[4] = {};

#if USE_TDM
  const unsigned int lds_base = (unsigned int)(size_t)(void*)lds_w;  // flat->LDS offset
  const unsigned int wave_off = (unsigned int)(wave * 16 * LDSTR * 4);
  // preload chunk 0 into buffer 0 (each wave DMAs its 16-row slice)
  tdm_load_w(W + (size_t)(wave * 16) * D + n0, lds_base + wave_off, D);
  __builtin_amdgcn_s_wait_tensorcnt(0);
  __syncthreads();

  int buf = 0;
  for (int c = 0; c < C; ++c) {
    if (c + 1 < C)  // prefetch next chunk into the other buffer (overlaps compute)
      tdm_load_w(W + (size_t)((c + 1) * KBLK + wave * 16) * D + n0,
                 lds_base + (unsigned int)((buf ^ 1) * BUF_FLOATS * 4) + wave_off, D);
    const float* lw = lds_w + buf * BUF_FLOATS;
    const float* xa = xrow + c * KBLK;
    for (int kk = 0; kk < KBLK; kk += 4) {
      v2f a = *(const v2f*)(xa + kk + koff);
#pragma unroll
      for (int t = 0; t < 4; ++t) {
        v2f b;
        b.x = lw[(kk + koff) * LDSTR + t * 16 + r];
        b.y = lw[(kk + koff + 1) * LDSTR + t * 16 + r];
        acc[t] = __builtin_amdgcn_wmma_f32_16x16x4_f32(
            false, a, false, b, (short)0, acc[t], false, false);
      }
    }
    __builtin_amdgcn_s_wait_tensorcnt(0);
    __syncthreads();
    buf ^= 1;
  }
#else
  // Fallback: cooperative global->LDS staging, single buffer.
  for (int c = 0; c < C; ++c) {
    const float* wc = W + (size_t)c * KBLK * D + n0;
    for (int i = threadIdx.x; i < KBLK * 16; i += 128) {  // 16 float4 per row
      int row = i >> 4, c4 = (i & 15) << 2;
      float4 v = *(const float4*)(wc + (size_t)row * D + c4);
      *(float4*)(lds_w + row * LDSTR + c4) = v;
    }
    __syncthreads();
    const float* xa = xrow + c * KBLK;
    for (int kk = 0; kk < KBLK; kk += 4) {
      v2f a = *(const v2f*)(xa + kk + koff);
#pragma unroll
      for (int t = 0; t < 4; ++t) {
        v2f b;
        b.x = lds_w[(kk + koff) * LDSTR + t * 16 + r];
        b.y = lds_w[(kk + koff + 1) * LDSTR + t * 16 + r];
        acc[t] = __builtin_amdgcn_wmma_f32_16x16x4_f32(
            false, a, false, b, (short)0, acc[t], false, false);
      }
    }
    __syncthreads();
  }
#endif

#pragma unroll
  for (int t = 0; t < 4; ++t) {
    float* out = H + (size_t)(mbase + (hi ? 8 : 0)) * D + n0 + t * 16 + r;
#pragma unroll
    for (int rr = 0; rr < 8; ++rr)
      out[(size_t)rr * D] = acc[t][rr];
  }
}

// ---------------------------------------------------------------------------
// Simple direct-global WMMA GEMM (used only for the tiny last layer, D=16).
// ---------------------------------------------------------------------------
template<int NT>
__global__ void gemm_wmma_f32(const float* __restrict__ X, const float* __restrict__ W,
                              float* __restrict__ H, int K, int D) {
  const int lane = threadIdx.x & 31;
  const int m0 = blockIdx.x << 4;
  const int n0 = blockIdx.y * (16 * NT);
  const int r = lane & 15;
  const bool hi = lane >= 16;
  const int koff = hi ? 2 : 0;

  v8f acc[NT] = {};
  const float* __restrict__ xrow = X + (size_t)(m0 + r) * K;

  for (int k0 = 0; k0 < K; k0 += 4) {
    v2f a = *(const v2f*)(xrow + k0 + koff);
    const float* wb = W + (size_t)(k0 + koff) * D + n0 + r;
#pragma unroll
    for (int t = 0; t < NT; ++t) {
      v2f b;
      b.x = wb[t * 16];
      b.y = wb[t * 16 + D];
      acc[t] = __builtin_amdgcn_wmma_f32_16x16x4_f32(
          false, a, false, b, (short)0, acc[t], false, false);
    }
  }
#pragma unroll
  for (int t = 0; t < NT; ++t) {
    float* out = H + (size_t)(m0 + (hi ? 8 : 0)) * D + n0 + t * 16 + r;
#pragma unroll
    for (int rr = 0; rr < 8; ++rr)
      out[(size_t)rr * D] = acc[t][rr];
  }
}

// es[m] = H[m,:].a_s ; ed[m] = H[m,:].a_d  — one wave per row, shuffle reduce.
__global__ void rowdot2(const float* __restrict__ H, const float* __restrict__ a_s,
                        const float* __restrict__ a_d, float* __restrict__ es,
                        float* __restrict__ ed, int D) {
  const int row = blockIdx.x;
  const int lane = threadIdx.x;
  const float* h = H + (size_t)row * D;
  float s0 = 0.f, s1 = 0.f;
  for (int j = lane; j < D; j += 32) {
    float v = h[j];
    s0 += v * a_s[j];
    s1 += v * a_d[j];
  }
#pragma unroll
  for (int off = 16; off > 0; off >>= 1) {
    s0 += __shfl_xor(s0, off, 32);
    s1 += __shfl_xor(s1, off, 32);
  }
  if (lane == 0) { es[row] = s0; ed[row] = s1; }
}

__global__ void init_nodes(int* __restrict__ m_ord, float* __restrict__ denom, int N) {
  int i = blockIdx.x * blockDim.x + threadIdx.x;
  if (i < N) {
    m_ord[i] = fenc(-INFINITY);
    denom[i] = 0.f;
  }
}

__global__ void zero_buf(float* __restrict__ p, size_t n) {
  size_t i = (size_t)blockIdx.x * blockDim.x + threadIdx.x;
  if (i < n) p[i] = 0.f;
}

// e = leaky_relu(es[src] + ed[dst]); segment max into m_ord[dst].
__global__ void edge_logit_max(const int* __restrict__ src, const int* __restrict__ dst,
                               const float* __restrict__ es, const float* __restrict__ ed,
                               float* __restrict__ e_edge, int* __restrict__ m_ord,
                               int E_BASE, int E_TOT) {
  int i = blockIdx.x * blockDim.x + threadIdx.x;
  if (i >= E_TOT) return;
  int s = (i < E_BASE) ? src[i] : (i - E_BASE);  // self-loops appended
  int d = (i < E_BASE) ? dst[i] : (i - E_BASE);
  float e = es[s] + ed[d];
  e = (e > 0.f) ? e : GAT_NEG_SLOPE * e;
  e_edge[i] = e;
  atomicMax(&m_ord[d], fenc(e));
}

// p = exp(e - m[dst]); segment sum into denom[dst].
__global__ void edge_exp_sum(const int* __restrict__ dst, const float* __restrict__ e_edge,
                             const int* __restrict__ m_ord, float* __restrict__ p_edge,
                             float* __restrict__ denom, int E_BASE, int E_TOT) {
  int i = blockIdx.x * blockDim.x + threadIdx.x;
  if (i >= E_TOT) return;
  int d = (i < E_BASE) ? dst[i] : (i - E_BASE);
  float m = fdec(m_ord[d]);
  if (!(m > -3.0e38f)) m = 0.f;  // mimic where(isfinite(m), m, 0)
  float p = expf(e_edge[i] - m);
  p_edge[i] = p;
  atomicAdd(&denom[d], p);
}

// out[dst,:] += (p/denom[dst]) * H[src,:]  — one block per edge.
__global__ void aggregate(const int* __restrict__ src, const int* __restrict__ dst,
                          const float* __restrict__ p_edge, const float* __restrict__ denom,
                          const float* __restrict__ H, float* __restrict__ out,
                          int D, int E_BASE, int E_TOT) {
  int i = blockIdx.x;
  if (i >= E_TOT) return;
  int s = (i < E_BASE) ? src[i] : (i - E_BASE);
  int d = (i < E_BASE) ? dst[i] : (i - E_BASE);
  float alpha = p_edge[i] / (denom[d] + 1e-16f);
  const float* hs = H + (size_t)s * D;
  float* od = out + (size_t)d * D;
  for (int f = threadIdx.x; f < D; f += blockDim.x)
    atomicAdd(&od[f], alpha * hs[f]);
}

__global__ void bias_act(float* __restrict__ out, const float* __restrict__ bias,
                         size_t n, int dmask, int relu) {
  size_t i = (size_t)blockIdx.x * blockDim.x + threadIdx.x;
  if (i >= n) return;
  float v = out[i] + bias[(int)(i & (size_t)dmask)];
  if (relu) v = fmaxf(v, 0.f);
  out[i] = v;
}

__global__ void log_softmax16(const float* __restrict__ in, float* __restrict__ out, int N) {
  int row = blockIdx.x * blockDim.x + threadIdx.x;
  if (row >= N) return;
  const float* r = in + (size_t)row * 16;
  float mx = r[0];
#pragma unroll
  for (int j = 1; j < 16; ++j) mx = fmaxf(mx, r[j]);
  float s = 0.f;
#pragma unroll
  for (int j = 0; j < 16; ++j) s += expf(r[j] - mx);
  float ls = mx + logf(s);
  float* o = out + (size_t)row * 16;
#pragma unroll
  for (int j = 0; j < 16; ++j) o[j] = r[j] - ls;
}

extern "C" void kernel_launch(void* const* d_in, const int* in_sizes, int n_in,
                              void* d_out, int out_size, void* d_ws, size_t ws_size,
                              hipStream_t stream) {
  const float* x = (const float*)d_in[0];
  const int* ei = (const int*)d_in[1];

  const int N = in_sizes[0] / 512;     // 10000
  const int E_BASE = in_sizes[1] / 2;  // 160000
  const int E_TOT = E_BASE + N;        // + self-loops
  const int* srcp = ei;
  const int* dstp = ei + E_BASE;

  const float *Wl[5], *As[5], *Ad[5], *Bs[5];
  for (int i = 0; i < 5; ++i) {
    Wl[i] = (const float*)d_in[2 + 4 * i];
    As[i] = (const float*)d_in[3 + 4 * i];
    Ad[i] = (const float*)d_in[4 + 4 * i];
    Bs[i] = (const float*)d_in[5 + 4 * i];
  }

  const int dims[6] = {512, 1024, 1024, 512, 256, 16};

  // Workspace carve (floats): two N x 1024 ping-pong buffers + small arrays.
  const size_t cap = (size_t)N * 1024;
  float* bufA = (float*)d_ws;  // h of current layer
  float* bufB = bufA + cap;    // activations (layer in/out, reused)
  float* es = bufB + cap;
  float* ed = es + N;
  float* denom = ed + N;
  int* m_ord = (int*)(denom + N);
  float* e_edge = (float*)(m_ord + N);
  float* p_edge = e_edge + E_TOT;

  const float* in = x;
  for (int l = 0; l < 5; ++l) {
    const int din = dims[l], dout = dims[l + 1];
    const size_t tot = (size_t)N * dout;

    // 1) h = in @ W  (reads `in`, writes bufA)
    if (dout % 64 == 0) {
      int gx = (N / 16 + 3) / 4;  // 64-row blocks, tail clamped in-kernel
      gemm_wmma_tile<<<dim3(gx, dout / 64), 128, 0, stream>>>(in, Wl[l], bufA, N, din, dout);
    } else {
      gemm_wmma_f32<1><<<dim3(N / 16, dout / 16), 32, 0, stream>>>(in, Wl[l], bufA, din, dout);
    }

    // 2) per-node attention logits
    rowdot2<<<N, 32, 0, stream>>>(bufA, As[l], Ad[l], es, ed, dout);

    // 3) reset segment state and output accumulator (bufB: safe — GEMM done reading it)
    init_nodes<<<(N + 255) / 256, 256, 0, stream>>>(m_ord, denom, N);
    zero_buf<<<(unsigned)((tot + 255) / 256), 256, 0, stream>>>(bufB, tot);

    // 4) edge logits + segment max / exp + segment sum
    edge_logit_max<<<(E_TOT + 255) / 256, 256, 0, stream>>>(srcp, dstp, es, ed, e_edge, m_ord,
                                                            E_BASE, E_TOT);
    edge_exp_sum<<<(E_TOT + 255) / 256, 256, 0, stream>>>(dstp, e_edge, m_ord, p_edge, denom,
                                                          E_BASE, E_TOT);

    // 5) weighted scatter aggregation
    int bt = (dout < 256) ? dout : 256;
    aggregate<<<E_TOT, bt, 0, stream>>>(srcp, dstp, p_edge, denom, bufA, bufB, dout, E_BASE,
                                        E_TOT);

    // 6) bias + ReLU (no ReLU after last layer)
    bias_act<<<(unsigned)((tot + 255) / 256), 256, 0, stream>>>(bufB, Bs[l], tot, dout - 1,
                                                                (l < 4) ? 1 : 0);
    in = bufB;
  }

  log_softmax16<<<(N + 255) / 256, 256, 0, stream>>>(bufB, (float*)d_out, N);
}